// ExpertsChooseMaskedContract_25348896981196
// MI455X (gfx1250) — compile-verified
//
#include <hip/hip_runtime.h>

#if defined(__HIP_DEVICE_COMPILE__)
#if !__has_builtin(__builtin_amdgcn_wmma_f32_16x16x4_f32)
#error "missing __builtin_amdgcn_wmma_f32_16x16x4_f32 for gfx1250 (device pass)"
#endif
#endif

typedef float v2f __attribute__((ext_vector_type(2)));
typedef float v8f __attribute__((ext_vector_type(8)));

namespace {
constexpr int kB = 4, kT = 4096, kI = 1024, kE = 8, kC = 512, kO = 1024;
constexpr int kWavesPerBlock = 8;
}

__device__ __forceinline__ v8f wmma_f32(v2f a, v2f b, v8f c) {
#if defined(__HIP_DEVICE_COMPILE__)
  // D = A(16x4, f32) * B(4x16, f32) + C(16x16, f32)
  return __builtin_amdgcn_wmma_f32_16x16x4_f32(
      /*neg_a=*/false, a, /*neg_b=*/false, b,
      /*c_mod=*/(short)0, c, /*reuse_a=*/false, /*reuse_b=*/false);
#else
  return c;
#endif
}

// ---------------------------------------------------------------------------
// Kernel 0: weight [E,O,I] -> wT [E,I,O] (LDS-tiled 32x32 transpose)
// ---------------------------------------------------------------------------
__global__ __launch_bounds__(256) void k_transpose_w(
    const float* __restrict__ w, float* __restrict__ wT) {
  __shared__ float tile[32][33];
  const int e = blockIdx.z;
  const int oBase = blockIdx.x * 32;
  const int iBase = blockIdx.y * 32;
  const int tx = threadIdx.x;            // 0..31
  const int ty = threadIdx.y;            // 0..7
  const float* we = w + e * kO * kI;
#pragma unroll
  for (int jj = 0; jj < 4; ++jj) {
    const int o = ty + 8 * jj;
    tile[o][tx] = we[(oBase + o) * kI + (iBase + tx)];
  }
  __syncthreads();
  float* wTe = wT + e * kI * kO;
#pragma unroll
  for (int jj = 0; jj < 4; ++jj) {
    const int i = ty + 8 * jj;
    wTe[(iBase + i) * kO + (oBase + tx)] = tile[tx][i];
  }
}

// ---------------------------------------------------------------------------
// Kernel 1: slotsT[b,e,i,c] = sum_t x[b,t,i] * mask[b,t,e,c]
//   GEMM per (b,e): M=I=1024 (rows i), N=C=512 (cols c), K=T=4096.
//   Wave tile: 64 (i) x 64 (c) = 4x4 WMMA subtiles -> 16 loads : 16 WMMAs.
// ---------------------------------------------------------------------------
__global__ __launch_bounds__(256) void k_stage1_slotsT(
    const float* __restrict__ x,     // [B,T,I]
    const float* __restrict__ mask,  // [B,T,E,C]
    float* __restrict__ slotsT) {    // [B,E,I,C]
  const int lane = threadIdx.x;      // 0..31
  const int h = lane >> 4;           // wave half
  const int r = lane & 15;
  const int gwave = blockIdx.x * kWavesPerBlock + threadIdx.y;
  const int be = gwave >> 7;                 // 32 (b,e) pairs
  const int tile = gwave & 127;              // 16 x 8 wave tiles
  const int iBase = (tile >> 3) * 64;        // I/64 = 16
  const int cBase = (tile & 7) * 64;         // C/64 = 8
  const int b = be >> 3;
  const int e = be & 7;

  const float* xb = x + b * kT * kI;
  const float* mbe = mask + b * kT * kE * kC + e * kC;
  constexpr int kMStride = kE * kC;          // mask stride per t (floats)

  const v8f vzero = {0.f, 0.f, 0.f, 0.f, 0.f, 0.f, 0.f, 0.f};
  v8f acc[4][4];
#pragma unroll
  for (int mt = 0; mt < 4; ++mt)
#pragma unroll
    for (int nt = 0; nt < 4; ++nt) acc[mt][nt] = vzero;

  for (int t = 0; t < kT; t += 4) {
    const int t0 = t + 2 * h;                // this half's K pair: t0, t0+1
    v2f afr[4], bfr[4];
#pragma unroll
    for (int mt = 0; mt < 4; ++mt) {         // A[m=i, k=t] = x[t][i]
      const int i = iBase + 16 * mt + r;
      afr[mt].x = xb[t0 * kI + i];
      afr[mt].y = xb[(t0 + 1) * kI + i];
    }
#pragma unroll
    for (int nt = 0; nt < 4; ++nt) {         // B[k=t, n=c] = mask[t][c]
      const int c = cBase + 16 * nt + r;
      bfr[nt].x = mbe[t0 * kMStride + c];
      bfr[nt].y = mbe[(t0 + 1) * kMStride + c];
    }
#pragma unroll
    for (int mt = 0; mt < 4; ++mt)
#pragma unroll
      for (int nt = 0; nt < 4; ++nt)
        acc[mt][nt] = wmma_f32(afr[mt], bfr[nt], acc[mt][nt]);
  }

  float* sb = slotsT + be * kI * kC;
#pragma unroll
  for (int mt = 0; mt < 4; ++mt)
#pragma unroll
    for (int nt = 0; nt < 4; ++nt) {
      const int c = cBase + 16 * nt + r;
#pragma unroll
      for (int j = 0; j < 8; ++j) {          // D row = j + 8*h
        const int i = iBase + 16 * mt + j + 8 * h;
        sb[i * kC + c] = acc[mt][nt][j];
      }
    }
}

// ---------------------------------------------------------------------------
// Kernel 2: out[b,e,c,o] = sum_i slotsT[b,e,i,c] * wT[e,i,o] + bias[e,o]
//   GEMM per (b,e): M=C=512 (rows c), N=O=1024 (cols o), K=I=1024.
//   Wave tile: 64 (c) x 64 (o) = 4x4 WMMA subtiles.
// ---------------------------------------------------------------------------
__global__ __launch_bounds__(256) void k_stage2_out(
    const float* __restrict__ slotsT,  // [B,E,I,C]
    const float* __restrict__ wT,      // [E,I,O]
    const float* __restrict__ bias,    // [E,O]
    float* __restrict__ out) {         // [B,E,C,O]
  const int lane = threadIdx.x;
  const int h = lane >> 4;
  const int r = lane & 15;
  const int gwave = blockIdx.x * kWavesPerBlock + threadIdx.y;
  const int be = gwave >> 7;                 // 32 (b,e) pairs
  const int tile = gwave & 127;              // 8 x 16 wave tiles
  const int cBase = (tile >> 4) * 64;        // C/64 = 8
  const int oBase = (tile & 15) * 64;        // O/64 = 16
  const int e = be & 7;

  const float* sbe = slotsT + be * kI * kC;
  const float* we = wT + e * kI * kO;

  const v8f vzero = {0.f, 0.f, 0.f, 0.f, 0.f, 0.f, 0.f, 0.f};
  v8f acc[4][4];
#pragma unroll
  for (int mt = 0; mt < 4; ++mt)
#pragma unroll
    for (int nt = 0; nt < 4; ++nt) acc[mt][nt] = vzero;

  for (int i = 0; i < kI; i += 4) {
    const int i0 = i + 2 * h;
    v2f afr[4], bfr[4];
#pragma unroll
    for (int mt = 0; mt < 4; ++mt) {         // A[m=c, k=i] = slotsT[i][c]
      const int c = cBase + 16 * mt + r;
      afr[mt].x = sbe[i0 * kC + c];
      afr[mt].y = sbe[(i0 + 1) * kC + c];
    }
#pragma unroll
    for (int nt = 0; nt < 4; ++nt) {         // B[k=i, n=o] = wT[i][o]
      const int o = oBase + 16 * nt + r;
      bfr[nt].x = we[i0 * kO + o];
      bfr[nt].y = we[(i0 + 1) * kO + o];
    }
#pragma unroll
    for (int mt = 0; mt < 4; ++mt)
#pragma unroll
      for (int nt = 0; nt < 4; ++nt)
        acc[mt][nt] = wmma_f32(afr[mt], bfr[nt], acc[mt][nt]);
  }

  float bv[4];
#pragma unroll
  for (int nt = 0; nt < 4; ++nt) bv[nt] = bias[e * kO + oBase + 16 * nt + r];

  float* ob = out + be * kC * kO;
#pragma unroll
  for (int mt = 0; mt < 4; ++mt)
#pragma unroll
    for (int nt = 0; nt < 4; ++nt) {
      const int o = oBase + 16 * nt + r;
#pragma unroll
      for (int j = 0; j < 8; ++j) {
        const int c = cBase + 16 * mt + j + 8 * h;
        ob[c * kO + o] = acc[mt][nt][j] + bv[nt];
      }
    }
}

// ---------------------------------------------------------------------------
extern "C" void kernel_launch(void* const* d_in, const int* in_sizes, int n_in,
                              void* d_out, int out_size, void* d_ws, size_t ws_size,
                              hipStream_t stream) {
  const float* x      = (const float*)d_in[0];  // [B,T,I]
  const float* mask   = (const float*)d_in[1];  // [B,T,E,C]
  const float* weight = (const float*)d_in[2];  // [E,O,I]
  const float* bias   = (const float*)d_in[3];  // [E,O]
  float* out = (float*)d_out;                   // [B,E,C,O]

  float* slotsT = (float*)d_ws;                               // B*E*I*C floats
  float* wT = slotsT + (size_t)kB * kE * kI * kC;             // E*I*O floats

  dim3 blk(32, 8);  // 8 wave32s per block

  // weight transpose: [E,O,I] -> [E,I,O]
  k_transpose_w<<<dim3(kO / 32, kI / 32, kE), blk, 0, stream>>>(weight, wT);

  // stage 1: 32 (b,e) * 128 wave tiles = 4096 waves -> 512 blocks
  k_stage1_slotsT<<<dim3(512), blk, 0, stream>>>(x, mask, slotsT);

  // stage 2: 32 (b,e) * 128 wave tiles = 4096 waves -> 512 blocks
  k_stage2_out<<<dim3(512), blk, 0, stream>>>(slotsT, wT, bias, out);
}